// RSNN_54099408060861
// MI455X (gfx1250) — compile-verified
//
#include <hip/hip_runtime.h>

// RSNN: xp = LN(x @ W_in^T + b_in)  (parallel GEMM, fp32 WMMA 16x16x4)
// then per-t recurrence: mem = 0.9*mem + xp_t + spk @ V_w^T + V_b - reset;
// spk = (mem>1); mem = LN(mem) -> output.  Recurrence is independent per
// batch row, so 4 persistent blocks (16 rows each) run all T steps with LDS
// state and no grid sync.  d_out is used in place: xp in, mem_rec out.
// fp32 WMMA keeps exact dtype fidelity with the reference (spike thresholds
// make the recurrence chaotic under weight quantization).

#define B_DIM 64
#define T_DIM 2048
#define I_DIM 256
#define H_DIM 512
#define LDSP  516   // padded LDS row stride (floats): 4*row mod 64 banks

typedef __attribute__((ext_vector_type(2))) float v2f;
typedef __attribute__((ext_vector_type(8))) float v8f;

__device__ __forceinline__ v8f wmma_f32(v2f a, v2f b, v8f c) {
  // D = A(16x4) * B(4x16) + C(16x16), fp32, wave32
  return __builtin_amdgcn_wmma_f32_16x16x4_f32(
      /*neg_a=*/false, a, /*neg_b=*/false, b,
      /*c_mod=*/(short)0, c, /*reuse_a=*/false, /*reuse_b=*/false);
}

// ---------------- Phase 1: input projection + LayerNorm ----------------
// grid = (B*T)/16 blocks of 256 threads (8 waves); block owns 16 rows x 512.
// wave w computes n-tiles 4w..4w+3 over K=256.
__global__ __launch_bounds__(256) void snn_proj_ln(
    const float* __restrict__ x,     // (B*T, 256)
    const float* __restrict__ W_in,  // (512, 256)
    const float* __restrict__ b_in,
    const float* __restrict__ ln_g,
    const float* __restrict__ ln_b,
    float* __restrict__ out)         // (B*T, 512)
{
  __shared__ float tile[16 * LDSP];
  __shared__ float red_s[16][16];
  __shared__ float red_q[16][16];
  __shared__ float row_mu[16];
  __shared__ float row_rs[16];

  const int  tid  = threadIdx.x;
  const int  wave = tid >> 5;
  const int  lane = tid & 31;
  const int  l16  = lane & 15;
  const int  lh   = lane >> 4;                 // half-wave: K/M split
  const long row0 = (long)blockIdx.x * 16;

  v8f acc[4];
#pragma unroll
  for (int j = 0; j < 4; ++j)
#pragma unroll
    for (int r = 0; r < 8; ++r) acc[j][r] = 0.0f;

  const float* ap = x + (row0 + l16) * I_DIM + 2 * lh;
  for (int k = 0; k < I_DIM; k += 4) {
    v2f a; a.x = ap[k]; a.y = ap[k + 1];
#pragma unroll
    for (int j = 0; j < 4; ++j) {
      const int n = (wave * 4 + j) * 16 + l16;
      const float* bp = W_in + (long)n * I_DIM + k + 2 * lh;
      v2f b; b.x = bp[0]; b.y = bp[1];
      acc[j] = wmma_f32(a, b, acc[j]);
    }
  }

  // bias + stash full 16x512 row-tile in LDS for the layernorm
#pragma unroll
  for (int j = 0; j < 4; ++j) {
    const int n = (wave * 4 + j) * 16 + l16;
    const float bias = b_in[n];
#pragma unroll
    for (int r = 0; r < 8; ++r)
      tile[(r + 8 * lh) * LDSP + n] = acc[j][r] + bias;
  }
  __syncthreads();

  { // partial sums: 16 rows x 16 segments x 32 elems
    const int row = tid >> 4, seg = tid & 15;
    const float* tr = tile + row * LDSP + seg * 32;
    float s = 0.f, q = 0.f;
#pragma unroll
    for (int e = 0; e < 32; ++e) { float v = tr[e]; s += v; q += v * v; }
    red_s[row][seg] = s; red_q[row][seg] = q;
  }
  __syncthreads();
  if (tid < 16) {
    float s = 0.f, q = 0.f;
#pragma unroll
    for (int i = 0; i < 16; ++i) { s += red_s[tid][i]; q += red_q[tid][i]; }
    const float mu  = s * (1.0f / H_DIM);
    const float var = q * (1.0f / H_DIM) - mu * mu;
    row_mu[tid] = mu;
    row_rs[tid] = rsqrtf(var + 1e-5f);
  }
  __syncthreads();
  {
    const int row = tid >> 4, seg = tid & 15;
    const float mu = row_mu[row], rs = row_rs[row];
    const float* tr = tile + row * LDSP + seg * 32;
    const float* gg = ln_g + seg * 32;
    const float* bb = ln_b + seg * 32;
    float* op = out + (row0 + row) * H_DIM + seg * 32;
#pragma unroll
    for (int e = 0; e < 32; ++e)
      op[e] = (tr[e] - mu) * rs * gg[e] + bb[e];
  }
}

// ---------------- Phase 2: persistent recurrence ----------------
// grid = 4 blocks of 512 threads (16 waves); block owns 16 batch rows for all
// T steps. spk/mem state in LDS; V_w stays L2-resident (1 MB << 192 MB).
// wave w computes n-tiles 2w, 2w+1 each step (K=512, fp32 WMMA).
// inp(t) is HBM-resident (268 MB stream > L2): preload it into registers
// before the WMMA chain and prefetch inp(t+1) a full step ahead.
__global__ __launch_bounds__(512) void snn_recurrent(
    const float* __restrict__ V_w,    // (512, 512)
    const float* __restrict__ V_b,
    const float* __restrict__ lnm_g,
    const float* __restrict__ lnm_b,
    float* __restrict__ io)           // (B, T, 512): xp in, mem_rec out
{
  __shared__ float spk [16 * LDSP];
  __shared__ float memv[16 * LDSP];
  __shared__ float red_s[16][32];
  __shared__ float red_q[16][32];
  __shared__ float row_mu[16];
  __shared__ float row_rs[16];

  const int tid  = threadIdx.x;
  const int wave = tid >> 5;         // 0..15
  const int lane = tid & 31;
  const int l16  = lane & 15;
  const int lh   = lane >> 4;
  const int b0   = blockIdx.x * 16;

  for (int i = tid; i < 16 * LDSP; i += 512) { spk[i] = 0.f; memv[i] = 0.f; }
  __syncthreads();

  const int n0 = (wave * 2 + 0) * 16 + l16;
  const int n1 = (wave * 2 + 1) * 16 + l16;
  const float* vw0 = V_w + (long)n0 * H_DIM + 2 * lh;
  const float* vw1 = V_w + (long)n1 * H_DIM + 2 * lh;
  const float vb0 = V_b[n0];
  const float vb1 = V_b[n1];
  const float* apb = spk + l16 * LDSP + 2 * lh;

  for (int t = 0; t < T_DIM; ++t) {
    // Preload inp(t) (HBM) and mem(t-1) (LDS) into registers: these loads
    // retire underneath the 128-iteration WMMA chain below.
    float inp0[8], inp1[8], mp0[8], mp1[8];
#pragma unroll
    for (int r = 0; r < 8; ++r) {
      const int  m    = r + 8 * lh;
      const long base = ((long)(b0 + m) * T_DIM + t) * H_DIM;
      inp0[r] = io[base + n0];
      inp1[r] = io[base + n1];
      mp0[r]  = memv[m * LDSP + n0];
      mp1[r]  = memv[m * LDSP + n1];
    }
    // Prefetch inp(t+1): lanes at n0 (2 KB row stride) touch every 128B line
    // of next step's slice (n1 = n0+16 floats shares the line).
    if (t + 1 < T_DIM) {
#pragma unroll
      for (int r = 0; r < 8; ++r) {
        const int m = r + 8 * lh;
        __builtin_prefetch(&io[((long)(b0 + m) * T_DIM + (t + 1)) * H_DIM + n0], 0, 3);
      }
    }

    // S = spk @ V_w^T : 16x512, this wave -> 2 tiles, K=512
    v8f acc0, acc1;
#pragma unroll
    for (int r = 0; r < 8; ++r) { acc0[r] = 0.0f; acc1[r] = 0.0f; }
    for (int k = 0; k < H_DIM; k += 4) {
      v2f a;  a.x  = apb[k];   a.y  = apb[k + 1];       // ds_load from spk
      v2f b0; b0.x = vw0[k];   b0.y = vw0[k + 1];
      v2f b1; b1.x = vw1[k];   b1.y = vw1[k + 1];
      acc0 = wmma_f32(a, b0, acc0);
      acc1 = wmma_f32(a, b1, acc1);
    }
    __syncthreads();   // everyone finished reading spk(t-1)

    // leak + input + recurrent + bias - reset; fire; stash mem_new
#pragma unroll
    for (int j = 0; j < 2; ++j) {
      const int   n  = j ? n1 : n0;
      const float vb = j ? vb1 : vb0;
#pragma unroll
      for (int r = 0; r < 8; ++r) {
        const int   m     = r + 8 * lh;
        const float s     = j ? acc1[r] : acc0[r];
        const float mprev = j ? mp1[r]  : mp0[r];       // LN'd mem from t-1
        const float inp   = j ? inp1[r] : inp0[r];
        const float reset = (mprev > 1.0f) ? 1.0f : 0.0f;   // THRESHOLD=1
        const float mnew  = 0.9f * mprev + inp + s + vb - reset;
        spk [m * LDSP + n] = (mnew > 1.0f) ? 1.0f : 0.0f;   // heaviside
        memv[m * LDSP + n] = mnew;
      }
    }
    __syncthreads();

    { // layernorm(mem_new): 16 rows x 32 segments x 16 elems
      const int row = tid >> 5, seg = tid & 31;
      const float* tr = memv + row * LDSP + seg * 16;
      float s = 0.f, q = 0.f;
#pragma unroll
      for (int e = 0; e < 16; ++e) { float v = tr[e]; s += v; q += v * v; }
      red_s[row][seg] = s; red_q[row][seg] = q;
    }
    __syncthreads();
    if (tid < 16) {
      float s = 0.f, q = 0.f;
#pragma unroll
      for (int i = 0; i < 32; ++i) { s += red_s[tid][i]; q += red_q[tid][i]; }
      const float mu  = s * (1.0f / H_DIM);
      const float var = q * (1.0f / H_DIM) - mu * mu;
      row_mu[tid] = mu;
      row_rs[tid] = rsqrtf(var + 1e-5f);
    }
    __syncthreads();
    { // normalize -> feedback state + output (in place over xp slot)
      const int row = tid >> 5, seg = tid & 31;
      const float mu = row_mu[row], rs = row_rs[row];
      float* tr = memv + row * LDSP + seg * 16;
      const float* gg = lnm_g + seg * 16;
      const float* bb = lnm_b + seg * 16;
      float* op = io + ((long)(b0 + row) * T_DIM + t) * H_DIM + seg * 16;
#pragma unroll
      for (int e = 0; e < 16; ++e) {
        const float v = (tr[e] - mu) * rs * gg[e] + bb[e];
        tr[e] = v;
        op[e] = v;
      }
    }
    __syncthreads();   // state visible before next step's GEMM/elementwise
  }
}

extern "C" void kernel_launch(void* const* d_in, const int* in_sizes, int n_in,
                              void* d_out, int out_size, void* d_ws, size_t ws_size,
                              hipStream_t stream) {
  (void)in_sizes; (void)n_in; (void)out_size; (void)d_ws; (void)ws_size;
  const float* x     = (const float*)d_in[0];
  const float* W_in  = (const float*)d_in[1];
  const float* b_in  = (const float*)d_in[2];
  const float* ln_g  = (const float*)d_in[3];
  const float* ln_b  = (const float*)d_in[4];
  const float* V_w   = (const float*)d_in[5];
  const float* V_b   = (const float*)d_in[6];
  const float* lnm_g = (const float*)d_in[7];
  const float* lnm_b = (const float*)d_in[8];
  float* out = (float*)d_out;

  // Phase 1: (B*T)/16 = 8192 row-tiles
  snn_proj_ln<<<(B_DIM * T_DIM) / 16, 256, 0, stream>>>(
      x, W_in, b_in, ln_g, ln_b, out);
  // Phase 2: 4 persistent blocks, 16 batch rows each
  snn_recurrent<<<B_DIM / 16, 512, 0, stream>>>(
      V_w, V_b, lnm_g, lnm_b, out);
}